// CenterLoss_50319836840503
// MI455X (gfx1250) — compile-verified
//
#include <hip/hip_runtime.h>

// CenterLoss for MI455X (gfx1250, wave32).
// d_out layout: [0] = loss, [1 .. C*D] = new_centers (row-major).
// d_ws  layout: ws[0] = loss accumulator, ws[1..C] = float counts.

typedef __attribute__((ext_vector_type(2))) float v2f;
typedef __attribute__((ext_vector_type(8))) float v8f;

// Exact fp32 wave32 sum via V_WMMA_F32_16X16X4_F32:
// A (16x4) holds each lane's p twice -> sum(A) = 2*sum_lanes(p).
// B = ones, so D[m][n] = rowsum_m(A) (B layout irrelevant).
// D layout: VGPR r, lanes 0-15 -> row r; lanes 16-31 -> row r+8.
// Summing the 8 D VGPRs per lane yields sum(rows 0-7) on lanes 0-15 and
// sum(rows 8-15) on lanes 16-31; one shfl_xor(16) combines -> 2*sum(p).
__device__ __forceinline__ float wave_sum32(float p) {
#if __has_builtin(__builtin_amdgcn_wmma_f32_16x16x4_f32)
    v2f a; a[0] = p;    a[1] = p;
    v2f b; b[0] = 1.0f; b[1] = 1.0f;
    v8f c = {0.f, 0.f, 0.f, 0.f, 0.f, 0.f, 0.f, 0.f};
    v8f d = __builtin_amdgcn_wmma_f32_16x16x4_f32(
        /*neg_a=*/false, a, /*neg_b=*/false, b,
        /*c_mod=*/(short)0, c, /*reuse_a=*/false, /*reuse_b=*/false);
    float t = d[0] + d[1] + d[2] + d[3] + d[4] + d[5] + d[6] + d[7];
    float u = __shfl_xor(t, 16, 32);
    return (t + u) * 0.5f;
#else
    for (int off = 16; off > 0; off >>= 1) p += __shfl_xor(p, off, 32);
    return p;
#endif
}

__global__ void cl_init(float* __restrict__ upd, float* __restrict__ ws,
                        long n_upd, int n_ws) {
    const long stride = (long)gridDim.x * blockDim.x;
    long i = (long)blockIdx.x * blockDim.x + threadIdx.x;
    for (long j = i; j < n_upd; j += stride) upd[j] = 0.0f;
    for (long j = i; j < n_ws;  j += stride) ws[j]  = 0.0f;
}

// One wave per row (D == 256). Row index kept scalar so labels[] becomes an
// SMEM load and the center-row gather uses a scalar base address.
__global__ void cl_main(const float* __restrict__ feat,
                        const int*   __restrict__ labels,
                        const float* __restrict__ centers,
                        float* __restrict__ upd,      // [C*256] accumulator (pre-zeroed)
                        float* __restrict__ counts,   // [C] float counts (pre-zeroed)
                        float* __restrict__ lossAcc,  // [1] (pre-zeroed)
                        int nrows, int nwaves) {
    const int lane = (int)(threadIdx.x & 31u);
    const int w = __builtin_amdgcn_readfirstlane(
        (int)((blockIdx.x * blockDim.x + threadIdx.x) >> 5));

    float p = 0.0f;
    for (int r = w; r < nrows; r += nwaves) {
        const int lab = labels[r];  // scalar (r is SGPR)

        const float4* fr = (const float4*)(feat    + (size_t)r   * 256);
        const float4* cr = (const float4*)(centers + (size_t)lab * 256);

        // speculative prefetch of the next feature row (global_prefetch_b8)
        if (r + nwaves < nrows)
            __builtin_prefetch(feat + (size_t)(r + nwaves) * 256 + lane * 8, 0, 0);

        // two fully-coalesced 512B requests per row for features & centers
        const float4 f0 = fr[lane];
        const float4 f1 = fr[lane + 32];
        const float4 c0 = cr[lane];
        const float4 c1 = cr[lane + 32];

        float4 e0, e1;  // diff = center - feature (matches reference sign)
        e0.x = c0.x - f0.x; e0.y = c0.y - f0.y; e0.z = c0.z - f0.z; e0.w = c0.w - f0.w;
        e1.x = c1.x - f1.x; e1.y = c1.y - f1.y; e1.z = c1.z - f1.z; e1.w = c1.w - f1.w;

        p += e0.x * e0.x + e0.y * e0.y + e0.z * e0.z + e0.w * e0.w
           + e1.x * e1.x + e1.y * e1.y + e1.z * e1.z + e1.w * e1.w;

        // scatter-add into L2-resident accumulator
        float* u = upd + (size_t)lab * 256 + lane * 4;
        atomicAdd(u + 0,   e0.x); atomicAdd(u + 1,   e0.y);
        atomicAdd(u + 2,   e0.z); atomicAdd(u + 3,   e0.w);
        atomicAdd(u + 128, e1.x); atomicAdd(u + 129, e1.y);
        atomicAdd(u + 130, e1.z); atomicAdd(u + 131, e1.w);

        if (lane == 0) atomicAdd(&counts[lab], 1.0f);
    }

    // once-per-wave reduction (EXEC all ones here: no divergence above scope)
    const float ws = wave_sum32(p);
    if (lane == 0) atomicAdd(lossAcc, ws);
}

__global__ void cl_final(const float* __restrict__ centers,
                         const float* __restrict__ counts,
                         const float* __restrict__ lossAcc,
                         float* __restrict__ out,   // out[0]=loss, out[1..]=new_centers
                         long cd, float lossScale) {
    const long stride = (long)gridDim.x * blockDim.x;
    const long i0 = (long)blockIdx.x * blockDim.x + threadIdx.x;
    if (i0 == 0) out[0] = lossAcc[0] * lossScale;
    for (long i = i0; i < cd; i += stride) {
        const float cnt = counts[i >> 8];            // D == 256
        out[1 + i] = centers[i] - out[1 + i] / (cnt + 1.0f);
    }
}

extern "C" void kernel_launch(void* const* d_in, const int* in_sizes, int n_in,
                              void* d_out, int out_size, void* d_ws, size_t ws_size,
                              hipStream_t stream) {
    const float* feat    = (const float*)d_in[0];  // [B, 256]
    const int*   labels  = (const int*)  d_in[1];  // [B]
    const float* centers = (const float*)d_in[2];  // [C, 256]

    const int  Bn = in_sizes[1];
    const int  Dn = in_sizes[0] / Bn;              // 256
    const long CD = (long)in_sizes[2];
    const int  Cn = (int)(CD / Dn);

    float* out     = (float*)d_out;
    float* upd     = out + 1;          // update accumulator, becomes new_centers in-place
    float* ws      = (float*)d_ws;
    float* lossAcc = ws;               // ws[0]
    float* counts  = ws + 1;           // ws[1..Cn]

    // 1) zero accumulators (d_out/d_ws are poisoned by the harness)
    cl_init<<<4096, 256, 0, stream>>>(upd, ws, CD, Cn + 1);

    // 2) fused gather + loss + scatter-add. 256 threads = 8 waves/block.
    const int blocks = 2048, threads = 256;
    const int nwaves = blocks * (threads / 32);
    cl_main<<<blocks, threads, 0, stream>>>(feat, labels, centers,
                                            upd, counts, lossAcc, Bn, nwaves);

    // 3) finalize: new_centers = centers - upd/(counts+1); write scalar loss
    cl_final<<<8192, 256, 0, stream>>>(centers, counts, lossAcc, out,
                                       CD, 0.5f / (float)Bn);
}